// Model_39548058862045
// MI455X (gfx1250) — compile-verified
//
#include <hip/hip_runtime.h>

typedef __attribute__((ext_vector_type(2))) float v2f;
typedef __attribute__((ext_vector_type(8))) float v8f;

#define B_N   16
#define CIN   16
#define L_LEN 8192
#define NBR   24
#define COUT  64
#define KW    3

// W_eff stored ci-paired: index = ((kk*8 + ci/2)*WP + c)*2 + (ci&1)
#define WP      80              // padded c-stride; 2*WP = 160 ≡ 32 (mod 64) -> bank-half split
#define W_ELEMS (3*8*WP*2)      // 3840 floats
// x tile stored ci-paired: index = ((ci/2)*XS + lo)*2 + (ci&1), lo in [0,130)
#define XS      144             // 2*XS = 288 ≡ 32 (mod 64) -> bank-half split between halves

__device__ __forceinline__ float coeffk(int k) {
  // COEFFS: 2.0 for k in [1,5],[7,11],[13,17]; else 1.0
  return ((k >= 1 && k <= 5) || (k >= 7 && k <= 11) || (k >= 13 && k <= 17)) ? 2.0f : 1.0f;
}

// Fold the 24 branches into one effective 64x16x3 filter + 64 bias, in WMMA-friendly layout.
__global__ void prep_weights(const float* __restrict__ W,
                             const float* __restrict__ bias,
                             float* __restrict__ ws) {
  int t = blockIdx.x * blockDim.x + threadIdx.x;
  if (t < W_ELEMS) {
    int r  = t & 1;
    int q  = t >> 1;
    int c  = q % WP;
    int cp = (q / WP) & 7;
    int kk = (q / WP) >> 3;
    float s = 0.0f;
    if (c < COUT) {
      int ci = cp * 2 + r;
      #pragma unroll
      for (int k = 0; k < NBR; ++k)
        s += coeffk(k) * W[((k * COUT + c) * CIN + ci) * KW + kk];
    }
    ws[t] = s;
  }
  if (t < COUT) {
    float s = 0.0f;
    #pragma unroll
    for (int k = 0; k < NBR; ++k) s += coeffk(k) * bias[k * COUT + t];
    ws[W_ELEMS + t] = s;
  }
}

// One block = one batch row x 128 contiguous l. 8 waves; each wave owns a 64(cout)x16(l) tile.
__global__ __launch_bounds__(256)
void conv_wmma(const float* __restrict__ x,
               const float* __restrict__ weff,
               float* __restrict__ out) {
  __shared__ float Wl[W_ELEMS];
  __shared__ float Xl[8 * XS * 2];
  __shared__ float Bl[COUT];

  const int tid   = threadIdx.x;
  const int b     = blockIdx.y;
  const int lbase = blockIdx.x * 128;

  // Stage effective weights + bias
  for (int i = tid; i < W_ELEMS; i += 256) Wl[i] = weff[i];
  if (tid < COUT) Bl[tid] = weff[W_ELEMS + tid];

  // Stage x tile with halo: lo=0 <-> global l = lbase-1; 130 positions, zero-padded edges
  for (int i = tid; i < CIN * 130; i += 256) {
    int ci = i / 130;
    int lo = i - ci * 130;
    int gl = lbase - 1 + lo;
    float v = 0.0f;
    if (gl >= 0 && gl < L_LEN) v = x[(b * CIN + ci) * L_LEN + gl];
    Xl[((ci >> 1) * XS + lo) * 2 + (ci & 1)] = v;
  }
  __syncthreads();

  const int wave = tid >> 5;
  const int lane = tid & 31;
  const int n    = lane & 15;   // N column (and M row for the A fragment)
  const int h    = lane >> 4;   // K-half selector (lanes 16..31 hold K=2,3)
  const int col0 = wave * 16;   // this wave's column base within the block

  v8f acc[4] = {};              // 4 M-tiles of 16x16 f32 accumulators

  #pragma unroll
  for (int kk = 0; kk < 3; ++kk) {          // conv tap
    #pragma unroll
    for (int ks = 0; ks < 4; ++ks) {        // K-step over cin (4 at a time)
      const int ci0 = ks * 4 + 2 * h;       // even
      const int cp  = ci0 >> 1;
      // B fragment: x[ci0, l+kk-1], x[ci0+1, l+kk-1] -> one ds_load_b64
      v2f bf = *(const v2f*)&Xl[(cp * XS + (col0 + n + kk)) * 2];
      #pragma unroll
      for (int mt = 0; mt < 4; ++mt) {
        // A fragment: W_eff[c, ci0, kk], W_eff[c, ci0+1, kk] -> one ds_load_b64
        v2f af = *(const v2f*)&Wl[((kk * 8 + cp) * WP + (mt * 16 + n)) * 2];
        acc[mt] = __builtin_amdgcn_wmma_f32_16x16x4_f32(
            false, af, false, bf, (short)0, acc[mt], false, false);
      }
    }
  }

  // Epilogue: + b_eff, ReLU, store. C/D layout: VGPR j -> M=j+8h, lane&15 -> N.
  const int l = lbase + col0 + n;
  #pragma unroll
  for (int mt = 0; mt < 4; ++mt) {
    #pragma unroll
    for (int j = 0; j < 8; ++j) {
      int c = mt * 16 + j + 8 * h;
      float v = acc[mt][j] + Bl[c];
      out[(b * COUT + c) * L_LEN + l] = fmaxf(v, 0.0f);
    }
  }
}

extern "C" void kernel_launch(void* const* d_in, const int* in_sizes, int n_in,
                              void* d_out, int out_size, void* d_ws, size_t ws_size,
                              hipStream_t stream) {
  const float* x    = (const float*)d_in[0];   // [16,16,8192]
  const float* W    = (const float*)d_in[1];   // [24,64,16,3]
  const float* bias = (const float*)d_in[2];   // [24,64]
  float* out = (float*)d_out;                  // [16,64,8192]
  float* ws  = (float*)d_ws;                   // >= (3840+64)*4 bytes

  prep_weights<<<dim3((W_ELEMS + 255) / 256), 256, 0, stream>>>(W, bias, ws);
  conv_wmma<<<dim3(L_LEN / 128, B_N), 256, 0, stream>>>(x, ws, out);
}